// PureSparseBackboneEPND2_Ex1_38079180046731
// MI455X (gfx1250) — compile-verified
//
#include <hip/hip_runtime.h>

// ---------------------------------------------------------------------------
// Problem constants (fixed by the reference file)
// ---------------------------------------------------------------------------
#define N1C 100000
#define N2C 80000
#define FC  32
#define K5C 125
#define K3C 27
#define M1C 20000
#define M2C 20000
#define M3C 30000
#define M4C 20000

typedef __attribute__((ext_vector_type(2))) float v2f;
typedef __attribute__((ext_vector_type(8))) float v8f;

// ---------------------------------------------------------------------------
// Utility kernels
// ---------------------------------------------------------------------------
__global__ void zero_kernel(float* __restrict__ p, size_t n) {
    size_t i = (size_t)blockIdx.x * blockDim.x + threadIdx.x;
    size_t s = (size_t)gridDim.x * blockDim.x;
    for (; i < n; i += s) p[i] = 0.0f;
}

__global__ void relu_kernel(float* __restrict__ p, size_t n) {
    size_t i = (size_t)blockIdx.x * blockDim.x + threadIdx.x;
    size_t s = (size_t)gridDim.x * blockDim.x;
    for (; i < n; i += s) p[i] = fmaxf(p[i], 0.0f);
}

// ---------------------------------------------------------------------------
// Layer 1a: Cin == 1 (rank-1 update) -> plain VALU + atomics, WMMA would be
// 97% K-padding. One thread per (k,m) pair; 32 contiguous atomic adds.
// ---------------------------------------------------------------------------
__global__ void conv1a_kernel(const float* __restrict__ x,
                              const float* __restrict__ w,      // [K,1,32]
                              const int* __restrict__ in_idx,   // [K*M]
                              const int* __restrict__ out_idx,  // [K*M]
                              float* out, int total, int M) {
    int pair = blockIdx.x * blockDim.x + threadIdx.x;
    if (pair >= total) return;
    int k = pair / M;
    float xv = x[in_idx[pair]];
    float* orow = out + (size_t)out_idx[pair] * FC;
    const float* wr = w + (size_t)k * FC;
#pragma unroll
    for (int c = 0; c < FC; ++c) atomicAdd(orow + c, xv * wr[c]);
}

// ---------------------------------------------------------------------------
// Generic gather-GEMM-scatter sparse conv using V_WMMA_F32_16X16X4_F32.
// One wave handles a (k, 16-row m-tile, 32-col cout-tile):
//   acc(16x32,f32) = gather(feats, in_idx)[16 x CIN] @ W[k][CIN x 32-slice]
// accumulated over CIN in steps of 4, then atomically scattered by out_idx.
//
// Fragment layouts (ISA 7.12.2, wave32):
//   A 16x4  : lane = Mrow + 16*(K/2), vgpr = K%2
//   B 4x16  : lane = Ncol + 16*(K/2), vgpr = K%2   (mirror of A)
//   C/D 16x16: vgpr v, lanes 0-15 -> M=v, lanes 16-31 -> M=v+8, N = lane%16
// Wave-uniform early-exit => EXEC all ones inside active waves (WMMA req).
// ---------------------------------------------------------------------------
template <int CIN, int COUT>
__global__ __launch_bounds__(256)
void conv_wmma_kernel(const float* __restrict__ feats,
                      const float* __restrict__ W,       // [K, CIN, COUT]
                      const int*  __restrict__ in_idx,   // [K, M]
                      const int*  __restrict__ out_idx,  // [K, M]
                      float* out, int M, int K) {
    const int wave = (int)((blockIdx.x * blockDim.x + threadIdx.x) >> 5);
    const int lane = threadIdx.x & 31;
    const int mtiles = M >> 4;
    const int ntiles = COUT >> 5;           // two 16-wide WMMA N-tiles per wave
    const int total  = K * mtiles * ntiles;
    if (wave >= total) return;              // wave-uniform: EXEC stays full

    const int k  = wave / (mtiles * ntiles);
    const int r  = wave % (mtiles * ntiles);
    const int m0 = (r / ntiles) << 4;
    const int n0 = (r % ntiles) << 5;

    const int half = lane >> 4;             // which half-wave (K/2 or M/8 bit)
    const int lm   = lane & 15;

    const int* kin  = in_idx  + (size_t)k * M + m0;
    const int* kout = out_idx + (size_t)k * M + m0;

    // A fragment source: gathered feature row for this lane's M-row.
    const int arow = kin[lm];
    const float* aptr = feats + (size_t)arow * CIN + 2 * half;
    // B fragment source: W[k][.][n0+lm (+16)]
    const float* bptr = W + (size_t)k * CIN * COUT + n0 + lm;

    v8f acc0 = {};
    v8f acc1 = {};
    for (int c = 0; c < CIN; c += 4) {
        v2f a = *(const v2f*)(aptr + c);            // A[M=lm][c+2h], [c+2h+1]
        const float* bc = bptr + (size_t)(c + 2 * half) * COUT;
        v2f b0; b0.x = bc[0];  b0.y = bc[COUT];      // B[c+2h][n], B[c+2h+1][n]
        v2f b1; b1.x = bc[16]; b1.y = bc[COUT + 16];
        acc0 = __builtin_amdgcn_wmma_f32_16x16x4_f32(
            false, a, false, b0, (short)0, acc0, false, false);
        acc1 = __builtin_amdgcn_wmma_f32_16x16x4_f32(
            false, a, false, b1, (short)0, acc1, false, false);
    }

    // Scatter-add: vgpr v holds D[M = v + 8*half][N = n0+lm].
#pragma unroll
    for (int v = 0; v < 8; ++v) {
        int orow = kout[v + 8 * half];
        float* op = out + (size_t)orow * COUT + n0 + lm;
        atomicAdd(op,      acc0[v]);
        atomicAdd(op + 16, acc1[v]);
    }
}

// ---------------------------------------------------------------------------
// BatchNorm pass 1: per-column sum / sum-of-squares.
// C divides 256; block-local LDS tree then one atomic per column per block.
// ---------------------------------------------------------------------------
__global__ __launch_bounds__(256)
void bnstats_kernel(const float* __restrict__ x, float* __restrict__ stats,
                    int N, int C) {
    __shared__ float sh[512];
    const int t   = threadIdx.x;
    const int col = t % C;
    const int rpb = blockDim.x / C;
    const int stride = gridDim.x * rpb;
    float s = 0.f, ss = 0.f;
    for (int row = blockIdx.x * rpb + t / C; row < N; row += stride) {
        float v = x[(size_t)row * C + col];
        s += v;
        ss = fmaf(v, v, ss);
    }
    sh[t] = s;
    sh[256 + t] = ss;
    __syncthreads();
    if (t < C) {
        float a = 0.f, b = 0.f;
        for (int j = 0; j < rpb; ++j) {
            a += sh[t + j * C];
            b += sh[256 + t + j * C];
        }
        atomicAdd(&stats[t], a);
        atomicAdd(&stats[C + t], b);
    }
}

// ---------------------------------------------------------------------------
// BatchNorm pass 2 (+optional ReLU), safe in place.
// var = E[x^2] - E[x]^2 (population, matches jnp var(0)).
// ---------------------------------------------------------------------------
__global__ void bnapply_kernel(const float* x, float* y,
                               const float* __restrict__ stats,
                               const float* __restrict__ g,
                               const float* __restrict__ b,
                               size_t total, int C, float invN, int doRelu) {
    size_t i = (size_t)blockIdx.x * blockDim.x + threadIdx.x;
    size_t s = (size_t)gridDim.x * blockDim.x;
    for (; i < total; i += s) {
        int col = (int)(i % (size_t)C);
        float m   = stats[col] * invN;
        float var = fmaf(-m, m, stats[C + col] * invN);
        float rr  = 1.0f / sqrtf(var + 1e-5f);
        float v   = fmaf((x[i] - m) * rr, g[col], b[col]);
        if (doRelu) v = fmaxf(v, 0.f);
        y[i] = v;
    }
}

// ---------------------------------------------------------------------------
// Host-side orchestration
// ---------------------------------------------------------------------------
static inline int covergrid(size_t n, int cap) {
    size_t b = (n + 255) / 256;
    if (b > (size_t)cap) b = cap;
    if (b == 0) b = 1;
    return (int)b;
}

extern "C" void kernel_launch(void* const* d_in, const int* in_sizes, int n_in,
                              void* d_out, int out_size, void* d_ws, size_t ws_size,
                              hipStream_t stream) {
    (void)in_sizes; (void)n_in; (void)out_size; (void)ws_size;

    const float* x     = (const float*)d_in[0];
    const float* w1a   = (const float*)d_in[1];
    const float* w1b   = (const float*)d_in[2];
    const float* w1c   = (const float*)d_in[3];
    const float* w2    = (const float*)d_in[4];
    const float* w3a   = (const float*)d_in[5];
    const float* w3b   = (const float*)d_in[6];
    const float* bn1b_g = (const float*)d_in[7];
    const float* bn1b_b = (const float*)d_in[8];
    const float* bn1c_g = (const float*)d_in[9];
    const float* bn1c_b = (const float*)d_in[10];
    const float* bn3a_g = (const float*)d_in[11];
    const float* bn3a_b = (const float*)d_in[12];
    const float* bn3b_g = (const float*)d_in[13];
    const float* bn3b_b = (const float*)d_in[14];
    const int* km1a_in  = (const int*)d_in[15];
    const int* km1a_out = (const int*)d_in[16];
    const int* km1b_in  = (const int*)d_in[17];
    const int* km1b_out = (const int*)d_in[18];
    const int* km1c_in  = (const int*)d_in[19];
    const int* km1c_out = (const int*)d_in[20];
    const int* km2_in   = (const int*)d_in[21];
    const int* km2_out  = (const int*)d_in[22];
    const int* km3a_in  = (const int*)d_in[23];
    const int* km3a_out = (const int*)d_in[24];
    const int* km3b_in  = (const int*)d_in[25];
    const int* km3b_out = (const int*)d_in[26];

    // Workspace layout (floats)
    float* ws  = (float*)d_ws;
    float* h1  = ws;                         // [N1, 32]
    float* t1  = ws + 3200000;               // [N1, 32]
    float* h2  = ws + 6400000;               // [N2, 64]
    float* t3  = ws + 11520000;              // [N2, 128]
    float* stats = ws + 21760000;            // up to 2*256 floats

    float* xe1 = (float*)d_out;              // [N1, 32]
    float* xe2 = (float*)d_out + (size_t)N1C * FC;   // [N2, 256]

    const size_t szH1 = (size_t)N1C * 32;
    const size_t szH2 = (size_t)N2C * 64;
    const size_t szT3 = (size_t)N2C * 128;
    const size_t szE2 = (size_t)N2C * 256;

    auto convBlocks = [](int K, int M, int Cout) {
        size_t waves = (size_t)K * (size_t)(M / 16) * (size_t)(Cout / 32);
        return (int)((waves + 7) / 8);       // 8 waves / 256-thread block
    };

    // ---- enc1 layer a: conv(1->32, k5) + ReLU ------------------------------
    zero_kernel<<<covergrid(szH1, 8192), 256, 0, stream>>>(h1, szH1);
    {
        int total = K5C * M1C;
        conv1a_kernel<<<(total + 255) / 256, 256, 0, stream>>>(
            x, w1a, km1a_in, km1a_out, h1, total, M1C);
    }
    relu_kernel<<<covergrid(szH1, 8192), 256, 0, stream>>>(h1, szH1);

    // ---- enc1 layer b: conv(32->32, k5) + BN + ReLU ------------------------
    zero_kernel<<<covergrid(szH1, 8192), 256, 0, stream>>>(t1, szH1);
    conv_wmma_kernel<32, 32><<<convBlocks(K5C, M2C, 32), 256, 0, stream>>>(
        h1, w1b, km1b_in, km1b_out, t1, M2C, K5C);
    zero_kernel<<<1, 256, 0, stream>>>(stats, 64);
    bnstats_kernel<<<512, 256, 0, stream>>>(t1, stats, N1C, 32);
    bnapply_kernel<<<covergrid(szH1, 8192), 256, 0, stream>>>(
        t1, h1, stats, bn1b_g, bn1b_b, szH1, 32, 1.0f / N1C, 1);

    // ---- enc1 layer c: conv(32->32, k5) + BN + ReLU -> x_e1 ----------------
    zero_kernel<<<covergrid(szH1, 8192), 256, 0, stream>>>(xe1, szH1);
    conv_wmma_kernel<32, 32><<<convBlocks(K5C, M2C, 32), 256, 0, stream>>>(
        h1, w1c, km1c_in, km1c_out, xe1, M2C, K5C);
    zero_kernel<<<1, 256, 0, stream>>>(stats, 64);
    bnstats_kernel<<<512, 256, 0, stream>>>(xe1, stats, N1C, 32);
    bnapply_kernel<<<covergrid(szH1, 8192), 256, 0, stream>>>(
        xe1, xe1, stats, bn1c_g, bn1c_b, szH1, 32, 1.0f / N1C, 1);

    // ---- conv2: conv(32->64, k3), no activation ----------------------------
    zero_kernel<<<covergrid(szH2, 8192), 256, 0, stream>>>(h2, szH2);
    conv_wmma_kernel<32, 64><<<convBlocks(K3C, M3C, 64), 256, 0, stream>>>(
        xe1, w2, km2_in, km2_out, h2, M3C, K3C);

    // ---- enc2 layer a: conv(64->128, k3) + BN + ReLU -----------------------
    zero_kernel<<<covergrid(szT3, 8192), 256, 0, stream>>>(t3, szT3);
    conv_wmma_kernel<64, 128><<<convBlocks(K3C, M4C, 128), 256, 0, stream>>>(
        h2, w3a, km3a_in, km3a_out, t3, M4C, K3C);
    zero_kernel<<<1, 256, 0, stream>>>(stats, 256);
    bnstats_kernel<<<512, 256, 0, stream>>>(t3, stats, N2C, 128);
    bnapply_kernel<<<covergrid(szT3, 8192), 256, 0, stream>>>(
        t3, t3, stats, bn3a_g, bn3a_b, szT3, 128, 1.0f / N2C, 1);

    // ---- enc2 layer b: conv(128->256, k3) + BN + ReLU -> x_e2 --------------
    zero_kernel<<<covergrid(szE2, 8192), 256, 0, stream>>>(xe2, szE2);
    conv_wmma_kernel<128, 256><<<convBlocks(K3C, M4C, 256), 256, 0, stream>>>(
        t3, w3b, km3b_in, km3b_out, xe2, M4C, K3C);
    zero_kernel<<<2, 256, 0, stream>>>(stats, 512);
    bnstats_kernel<<<512, 256, 0, stream>>>(xe2, stats, N2C, 256);
    bnapply_kernel<<<covergrid(szE2, 8192), 256, 0, stream>>>(
        xe2, xe2, stats, bn3b_g, bn3b_b, szE2, 256, 1.0f / N2C, 1);
}